// MODEL_3298534883413
// MI455X (gfx1250) — compile-verified
//
#include <hip/hip_runtime.h>
#include <math.h>

// Problem constants (from reference)
#define B_   256
#define S_   512
#define NQ_  1000
#define MEM_ 20
#define DK_  50
#define DV_  200
#define FC_  50

typedef __attribute__((ext_vector_type(2))) float v2f;
typedef __attribute__((ext_vector_type(8))) float v8f;

// ---------------------------------------------------------------------------
// Kernel A1: corr_table[1001,20] = softmax(q_embed_w @ memory_key^T, axis=1)
// One block (1 wave) per question id.
// ---------------------------------------------------------------------------
__global__ void corr_table_kernel(const float* __restrict__ q_embed_w,
                                  const float* __restrict__ memory_key,
                                  float* __restrict__ corr_table) {
  const int row = blockIdx.x;          // 0..NQ_
  const int m   = threadIdx.x;         // 0..31
  __shared__ float lg[MEM_];
  __shared__ float smax, sinv;
  if (m < MEM_) {
    float acc = 0.0f;
    for (int k = 0; k < DK_; ++k)
      acc += q_embed_w[row * DK_ + k] * memory_key[m * DK_ + k];
    lg[m] = acc;
  }
  __syncthreads();
  if (m == 0) {
    float mx = lg[0];
    for (int i = 1; i < MEM_; ++i) mx = fmaxf(mx, lg[i]);
    float s = 0.0f;
    for (int i = 0; i < MEM_; ++i) s += expf(lg[i] - mx);
    smax = mx; sinv = 1.0f / s;
  }
  __syncthreads();
  if (m < MEM_) corr_table[row * MEM_ + m] = expf(lg[m] - smax) * sinv;
}

// ---------------------------------------------------------------------------
// Kernel A2: qproj_table[1001,50] = q_embed_w @ read_w[200:250,:] + read_b
// ---------------------------------------------------------------------------
__global__ void qproj_kernel(const float* __restrict__ q_embed_w,
                             const float* __restrict__ read_w,
                             const float* __restrict__ read_b,
                             float* __restrict__ qproj) {
  const int row = blockIdx.x;
  const int j   = threadIdx.x;
  if (j >= FC_) return;
  float acc = read_b[j];
  for (int k = 0; k < DK_; ++k)
    acc += q_embed_w[row * DK_ + k] * read_w[(DV_ + k) * FC_ + j];
  qproj[row * FC_ + j] = acc;
}

// ---------------------------------------------------------------------------
// Kernel A3: erase_table = sigmoid(qa_embed_w @ erase_w + erase_b)
//            add_table   = tanh   (qa_embed_w @ add_w   + add_b)
// (2001,200)@(200,200): fp32 WMMA 16x16x4, one 16x16 output tile per wave,
// A fragment shared between the two products. 126 row-tiles x 13 col-tiles.
// Bounds handled by CLAMPING load indices (garbage only reaches never-stored
// D rows/cols) so the inner loop has zero predication / exec manipulation.
// ---------------------------------------------------------------------------
__global__ void ea_table_kernel(const float* __restrict__ qa_w,
                                const float* __restrict__ erase_w,
                                const float* __restrict__ erase_b,
                                const float* __restrict__ add_w,
                                const float* __restrict__ add_b,
                                float* __restrict__ erase_t,
                                float* __restrict__ add_t) {
  const int NROWS = 2 * NQ_ + 1;             // 2001
  const int nNT   = (DV_ + 15) / 16;         // 13
  const int nRT   = (NROWS + 15) / 16;       // 126
  const int wave  = threadIdx.x >> 5;
  const int lane  = threadIdx.x & 31;
  const int task  = blockIdx.x * 8 + wave;   // one 16x16 tile pair per wave
  if (task >= nRT * nNT) return;             // wave-uniform exit (EXEC stays full)
  const int rt = task / nNT, nt = task % nNT;
  const int half = lane >> 4, l16 = lane & 15;
  const int arow = min(rt * 16 + l16, NROWS - 1);  // clamped A row (M)
  const int col  = min(nt * 16 + l16, DV_ - 1);    // clamped B/D column (N)
  const float* __restrict__ aptr = qa_w + arow * DV_;

  v8f de = {};
  v8f da = {};
#pragma unroll 5
  for (int kk = 0; kk < DV_ / 4; ++kk) {
    const int k0 = kk * 4 + half * 2;        // ISA A layout: lanes 0-15 K=0,1; 16-31 K=2,3
    const v2f a = *(const v2f*)(aptr + k0);  // 8B-aligned (k0 even) -> b64 load
    v2f be, ba;
    be.x = erase_w[(k0    ) * DV_ + col];
    be.y = erase_w[(k0 + 1) * DV_ + col];
    ba.x = add_w[(k0    ) * DV_ + col];
    ba.y = add_w[(k0 + 1) * DV_ + col];
    de = __builtin_amdgcn_wmma_f32_16x16x4_f32(false, a, false, be, (short)0, de, false, false);
    da = __builtin_amdgcn_wmma_f32_16x16x4_f32(false, a, false, ba, (short)0, da, false, false);
  }
  const float eb = erase_b[col];
  const float ab = add_b[col];
  const bool cok = (nt * 16 + l16) < DV_;
#pragma unroll
  for (int r = 0; r < 8; ++r) {
    const int orow = rt * 16 + half * 8 + r; // D layout: VGPR r -> M=r (+8 upper half)
    if (orow < NROWS && cok) {
      erase_t[orow * DV_ + col] = 1.0f / (1.0f + expf(-(de[r] + eb)));
      add_t  [orow * DV_ + col] = tanhf(da[r] + ab);
    }
  }
}

// ---------------------------------------------------------------------------
// Kernel B: the sequential scan. One workgroup (256 thr = 8 waves) per batch b.
// Mv state (20x200) lives in registers: lane d<200 holds Mv[:,d] (20 VGPRs).
// Every 16 steps, the buffered reads tile (16x200) is projected through
// read_w[0:200,:] (200x50) with fp32 WMMA, fused with tanh / pred head / BCE.
// Idle lanes prefetch the next step's gate-table rows (global_prefetch_b8).
// ---------------------------------------------------------------------------
#define RT_STRIDE 202   // even (8B-aligned b64 LDS loads); 10*l mod 64 distinct -> no conflicts
#define HT_STRIDE 52

__global__ void scan_kernel(const int*   __restrict__ q_data,
                            const int*   __restrict__ qa_data,
                            const float* __restrict__ target,
                            const float* __restrict__ init_mv,
                            const float* __restrict__ read_w,
                            const float* __restrict__ pred_w,
                            const float* __restrict__ pred_b,
                            const float* __restrict__ corr_table,
                            const float* __restrict__ qproj_table,
                            const float* __restrict__ erase_table,
                            const float* __restrict__ add_table,
                            float* __restrict__ out,      // d_out: [0]=loss, [1..]=sigmoid
                            float* __restrict__ accum) {  // [0]=bce sum, [1]=mask count
  const int b    = blockIdx.x;
  const int tid  = threadIdx.x;
  const int lane = tid & 31;
  const int wave = tid >> 5;
  const int l16  = lane & 15;
  const int half = lane >> 4;

  __shared__ float reads_tile[16 * RT_STRIDE];
  __shared__ float qproj_tile[16 * HT_STRIDE];
  __shared__ float h_tile[16 * HT_STRIDE];
  __shared__ float corr_s[MEM_];
  __shared__ float red_b[16], red_c[16];

  float mv[MEM_];
  if (tid < DV_) {
#pragma unroll
    for (int m = 0; m < MEM_; ++m) mv[m] = init_mv[m * DV_ + tid];
  }
  float bce_loc = 0.0f, cnt_loc = 0.0f;

  for (int t = 0; t < S_; ++t) {
    const int qid  = q_data[b * S_ + t];
    const int qaid = qa_data[b * S_ + t];
    const int trow = t & 15;
    if (tid < MEM_) corr_s[tid] = corr_table[qid * MEM_ + tid];
    if (tid >= DV_ && tid < DV_ + FC_)                    // idle lanes fetch qproj row
      qproj_tile[trow * HT_STRIDE + (tid - DV_)] = qproj_table[qid * FC_ + (tid - DV_)];
    if (t + 1 < S_) {                                     // prefetch next gate rows
      if (tid >= 224 && tid < 231) {
        const int qan = qa_data[b * S_ + t + 1];
        __builtin_prefetch(&erase_table[qan * DV_ + (tid - 224) * 32], 0, 3);
      } else if (tid >= 232 && tid < 239) {
        const int qan = qa_data[b * S_ + t + 1];
        __builtin_prefetch(&add_table[qan * DV_ + (tid - 232) * 32], 0, 3);
      }
    }
    __syncthreads();

    if (tid < DV_) {                                      // read + memory update, fused
      const float e = erase_table[qaid * DV_ + tid];
      const float a = add_table[qaid * DV_ + tid];
      float rd = 0.0f;
#pragma unroll
      for (int m = 0; m < MEM_; ++m) {
        const float w = corr_s[m];
        const float v = mv[m];
        rd += w * v;                                      // read uses pre-update Mv
        mv[m] = v * (1.0f - w * e) + w * a;
      }
      reads_tile[trow * RT_STRIDE + tid] = rd;
    }
    __syncthreads();

    if (trow == 15) {
      // h = tanh(reads(16x200) @ read_w_top(200x50) + qproj); waves 0..3 own n-tiles
      if (wave < 4) {
        const int col  = min(wave * 16 + l16, FC_ - 1);   // clamped load column
        const bool cok = (wave * 16 + l16) < FC_;
        v8f d = {};
#pragma unroll 5
        for (int kk = 0; kk < DV_ / 4; ++kk) {
          const int k0 = kk * 4 + half * 2;
          const v2f a = *(const v2f*)(&reads_tile[l16 * RT_STRIDE + k0]); // b64 ds load
          v2f bb;
          bb.x = read_w[(k0    ) * FC_ + col];
          bb.y = read_w[(k0 + 1) * FC_ + col];
          d = __builtin_amdgcn_wmma_f32_16x16x4_f32(false, a, false, bb, (short)0, d, false, false);
        }
        if (cok) {
#pragma unroll
          for (int r = 0; r < 8; ++r) {
            const int tl = half * 8 + r;
            h_tile[tl * HT_STRIDE + col] = tanhf(d[r] + qproj_tile[tl * HT_STRIDE + col]);
          }
        }
      }
      __syncthreads();
      if (tid < 16) {                                     // pred head + BCE per timestep
        float z = pred_b[0];
        for (int j = 0; j < FC_; ++j) z += h_tile[tid * HT_STRIDE + j] * pred_w[j];
        const int gt  = (t - 15) + tid;
        const int idx = b * S_ + gt;
        out[1 + idx] = 1.0f / (1.0f + expf(-z));
        const float tg = target[idx];
        if (tg >= 0.0f) {
          bce_loc += fmaxf(z, 0.0f) - z * tg + log1pf(expf(-fabsf(z)));
          cnt_loc += 1.0f;
        }
      }
      __syncthreads();
    }
  }

  if (tid < 16) { red_b[tid] = bce_loc; red_c[tid] = cnt_loc; }
  __syncthreads();
  if (tid == 0) {
    float sb = 0.0f, sc = 0.0f;
    for (int i = 0; i < 16; ++i) { sb += red_b[i]; sc += red_c[i]; }
    atomicAdd(&accum[0], sb);
    atomicAdd(&accum[1], sc);
  }
}

__global__ void zero_accum_kernel(float* __restrict__ accum) {
  if (threadIdx.x < 2) accum[threadIdx.x] = 0.0f;
}

__global__ void finalize_kernel(const float* __restrict__ accum, float* __restrict__ out) {
  out[0] = accum[0] / fmaxf(accum[1], 1.0f);
}

// ---------------------------------------------------------------------------
// Launcher
// ---------------------------------------------------------------------------
extern "C" void kernel_launch(void* const* d_in, const int* in_sizes, int n_in,
                              void* d_out, int out_size, void* d_ws, size_t ws_size,
                              hipStream_t stream) {
  const int*   q_data    = (const int*)  d_in[0];
  const int*   qa_data   = (const int*)  d_in[1];
  const float* target    = (const float*)d_in[2];
  const float* q_embed_w = (const float*)d_in[3];
  const float* qa_embed_w= (const float*)d_in[4];
  const float* memory_key= (const float*)d_in[5];
  const float* init_mv   = (const float*)d_in[6];
  const float* erase_w   = (const float*)d_in[7];
  const float* erase_b   = (const float*)d_in[8];
  const float* add_w     = (const float*)d_in[9];
  const float* add_b     = (const float*)d_in[10];
  const float* read_w    = (const float*)d_in[11];
  const float* read_b    = (const float*)d_in[12];
  const float* pred_w    = (const float*)d_in[13];
  const float* pred_b    = (const float*)d_in[14];
  float* out = (float*)d_out;

  // Workspace layout (floats); total ~3.5 MB
  float* ws = (float*)d_ws;
  float* corr_table  = ws;                                   // 1001*20   = 20020
  float* qproj_table = corr_table  + (NQ_ + 1) * MEM_;       // 1001*50   = 50050
  float* erase_table = qproj_table + (NQ_ + 1) * FC_;        // 2001*200  = 400200
  float* add_table   = erase_table + (2 * NQ_ + 1) * DV_;    // 2001*200  = 400200
  float* accum       = add_table   + (2 * NQ_ + 1) * DV_;    // 2

  zero_accum_kernel<<<1, 32, 0, stream>>>(accum);
  corr_table_kernel<<<NQ_ + 1, 32, 0, stream>>>(q_embed_w, memory_key, corr_table);
  qproj_kernel<<<NQ_ + 1, 64, 0, stream>>>(q_embed_w, read_w, read_b, qproj_table);

  const int nTiles = ((2 * NQ_ + 1 + 15) / 16) * ((DV_ + 15) / 16);  // 126*13 = 1638 waves
  ea_table_kernel<<<(nTiles + 7) / 8, 256, 0, stream>>>(qa_embed_w, erase_w, erase_b,
                                                        add_w, add_b,
                                                        erase_table, add_table);

  scan_kernel<<<B_, 256, 0, stream>>>(q_data, qa_data, target, init_mv, read_w,
                                      pred_w, pred_b,
                                      corr_table, qproj_table, erase_table, add_table,
                                      out, accum);

  finalize_kernel<<<1, 1, 0, stream>>>(accum, out);
}